// QCycleConv2d_24111946399869
// MI455X (gfx1250) — compile-verified
//
#include <hip/hip_runtime.h>
#include <math.h>

typedef __attribute__((ext_vector_type(2))) float v2f;
typedef __attribute__((ext_vector_type(8))) float v8f;

#define NIN   1024
#define NBIG  2048
#define BATCH 8
#define PI_F  3.14159265358979323846f
#define APITCH 17   // LDS pitch (dwords) for [kk][row16] A-tiles: conflict-free fill & read

// ---------------- Stage 0: build u = kron(ux,uy), uc, and weight table ----------------
__global__ __launch_bounds__(256) void stage0(const float* __restrict__ phi,
                                              float* __restrict__ ur, float* __restrict__ ui,
                                              float* __restrict__ wr, float* __restrict__ wi) {
  __shared__ float uxr[256], uxi[256], uyr[256], uyi[256];
  __shared__ float ucr[64], uci[64];
  const int t = threadIdx.x;
  {
    // ux[a,b] = (1/16) sum_k cis(phi_k + 2*pi*k*(a-b)/16); same for uy with phi[16..]
    int a = t >> 4, b = t & 15;
    float sxr = 0.f, sxi = 0.f, syr = 0.f, syi = 0.f;
    for (int k = 0; k < 16; ++k) {
      float ang = (2.0f * PI_F) * (float)(k * (a - b)) * (1.0f / 16.0f);
      float s, c;
      __sincosf(phi[k] + ang, &s, &c);       sxr += c; sxi += s;
      __sincosf(phi[16 + k] + ang, &s, &c);  syr += c; syi += s;
    }
    uxr[t] = sxr * 0.0625f; uxi[t] = sxi * 0.0625f;
    uyr[t] = syr * 0.0625f; uyi[t] = syi * 0.0625f;
  }
  __syncthreads();
  {
    // u[(a1*16+a2),(b1*16+b2)] = ux[a1,b1]*uy[a2,b2]; thread t owns row t
    int a1 = t >> 4, a2 = t & 15;
    for (int c = 0; c < 256; ++c) {
      int b1 = c >> 4, b2 = c & 15;
      float xr = uxr[a1 * 16 + b1], xi = uxi[a1 * 16 + b1];
      float yr = uyr[a2 * 16 + b2], yi = uyi[a2 * 16 + b2];
      ur[t * 256 + c] = xr * yr - xi * yi;
      ui[t * 256 + c] = xr * yi + xi * yr;
    }
  }
  if (t == 0) {
    // uc: 8-mode Clements interferometer, 28 MZIs, phases phi[32..87]
    float Ur[8][8] = {}, Ui[8][8] = {};
    for (int d = 0; d < 8; ++d) Ur[d][d] = 1.0f;
    int idx = 32;
    for (int layer = 0; layer < 8; ++layer) {
      for (int m0 = (layer & 1); m0 < 7; m0 += 2) {
        float s1, c1, s2, c2;
        __sincosf(phi[idx], &s1, &c1);
        __sincosf(phi[idx + 1], &s2, &c2);
        idx += 2;
        // M = 0.5 * [[e1*(e2+1), e2-1], [e1*(e2-1), e2+1]]
        float pr = c2 + 1.0f, pi_ = s2, mr = c2 - 1.0f, mi = s2;
        float M00r = 0.5f * (c1 * pr - s1 * pi_), M00i = 0.5f * (c1 * pi_ + s1 * pr);
        float M01r = 0.5f * mr,                   M01i = 0.5f * mi;
        float M10r = 0.5f * (c1 * mr - s1 * mi),  M10i = 0.5f * (c1 * mi + s1 * mr);
        float M11r = 0.5f * pr,                   M11i = 0.5f * pi_;
        for (int cc = 0; cc < 8; ++cc) {
          float r0r = Ur[m0][cc], r0i = Ui[m0][cc];
          float r1r = Ur[m0 + 1][cc], r1i = Ui[m0 + 1][cc];
          Ur[m0][cc]     = M00r * r0r - M00i * r0i + M01r * r1r - M01i * r1i;
          Ui[m0][cc]     = M00r * r0i + M00i * r0r + M01r * r1i + M01i * r1r;
          Ur[m0 + 1][cc] = M10r * r0r - M10i * r0i + M11r * r1r - M11i * r1i;
          Ui[m0 + 1][cc] = M10r * r0i + M10i * r0r + M11r * r1i + M11i * r1r;
        }
      }
    }
    for (int e = 0; e < 64; ++e) { ucr[e] = Ur[e >> 3][e & 7]; uci[e] = Ui[e >> 3][e & 7]; }
  }
  __syncthreads();
  // w[a][c][p][q] = uc[a][p+4] * conj(uc[c][q+4]); 1024 complex entries
  for (int s = 0; s < 4; ++s) {
    int n = t * 4 + s;
    int a = n >> 7, c = (n >> 4) & 7, p = (n >> 2) & 3, q = n & 3;
    float uar = ucr[a * 8 + p + 4], uai = uci[a * 8 + p + 4];
    float ubr = ucr[c * 8 + q + 4], ubi = uci[c * 8 + q + 4];
    wr[n] = uar * ubr + uai * ubi;
    wi[n] = uai * ubr - uar * ubi;
  }
}

// ---------------- Stage A: S = (I4 (x) u) * rho   (complex, per batch) ----------------
// Block = 8 waves sharing one 16-row A-tile of u (staged in LDS, [kk][row] pitch 17,
// conflict-free). Each wave owns one 16x16 output tile; rho streamed NT (read-once).
__global__ __launch_bounds__(256) void stageA(const float* __restrict__ rr, const float* __restrict__ ri,
                                              const float* __restrict__ ur, const float* __restrict__ ui,
                                              float* __restrict__ Sr, float* __restrict__ Si) {
  __shared__ float lAr[256 * APITCH];
  __shared__ float lAi[256 * APITCH];
  const int t    = threadIdx.x;
  const int lane = t & 31;
  const int wave = t >> 5;
  const int blk  = blockIdx.x;                 // 4096 = 8 batches * 64 ti * 8 tj-groups
  const int b    = blk >> 9;
  const int ti   = (blk >> 3) & 63;
  const int tjg  = blk & 7;
  const int tj   = tjg * 8 + wave;
  const int m  = lane & 15, n = lane & 15, hi = lane >> 4;
  const int p     = ti >> 4;                   // input row-block (channel)
  const int urowb = (ti & 15) * 16;            // first u row of this tile
  const int col   = tj * 16 + n;

  // cooperative fill of the shared A tile: row=iter, kk=t  (global coalesced, LDS stride-17)
#pragma unroll
  for (int iter = 0; iter < 16; ++iter) {
    lAr[t * APITCH + iter] = ur[(urowb + iter) * 256 + t];
    lAi[t * APITCH + iter] = ui[(urowb + iter) * 256 + t];
  }
  __syncthreads();

  const float* Rr = rr + (size_t)b * NIN * NIN;
  const float* Ri = ri + (size_t)b * NIN * NIN;
  v8f accr = {}, acci = {};
  for (int k = 0; k < 256; k += 4) {
    const int kk = k + 2 * hi;
    v2f ar, ai, nai;
    ar.x = lAr[kk * APITCH + m];       ar.y = lAr[(kk + 1) * APITCH + m];
    ai.x = lAi[kk * APITCH + m];       ai.y = lAi[(kk + 1) * APITCH + m];
    nai.x = -ai.x;                     nai.y = -ai.y;
    v2f br, bi;  // rho: read-once stream -> non-temporal
    br.x = __builtin_nontemporal_load(&Rr[(size_t)(p * 256 + kk)     * NIN + col]);
    br.y = __builtin_nontemporal_load(&Rr[(size_t)(p * 256 + kk + 1) * NIN + col]);
    bi.x = __builtin_nontemporal_load(&Ri[(size_t)(p * 256 + kk)     * NIN + col]);
    bi.y = __builtin_nontemporal_load(&Ri[(size_t)(p * 256 + kk + 1) * NIN + col]);
    accr = __builtin_amdgcn_wmma_f32_16x16x4_f32(false, ar,  false, br, (short)0, accr, false, false);
    accr = __builtin_amdgcn_wmma_f32_16x16x4_f32(false, nai, false, bi, (short)0, accr, false, false);
    acci = __builtin_amdgcn_wmma_f32_16x16x4_f32(false, ar,  false, bi, (short)0, acci, false, false);
    acci = __builtin_amdgcn_wmma_f32_16x16x4_f32(false, ai,  false, br, (short)0, acci, false, false);
  }
  float* sr = Sr + (size_t)b * NIN * NIN;
  float* si = Si + (size_t)b * NIN * NIN;
  const int rbase = ti * 16 + 8 * hi;
#pragma unroll
  for (int v = 0; v < 8; ++v) {
    sr[(size_t)(rbase + v) * NIN + col] = accr[v];
    si[(size_t)(rbase + v) * NIN + col] = acci[v];
  }
}

// ---------------- Stage B: T = S * (I4 (x) u)^H ----------------
// Tr = Sr*ur^T + Si*ui^T ; Ti = Si*ur^T - Sr*ui^T.
// A operand (16 rows of S, col-block q) staged in LDS per block; u rows read from
// global (512KB, L2-resident) as contiguous v2f per lane.
__global__ __launch_bounds__(256) void stageB(const float* __restrict__ Sr, const float* __restrict__ Si,
                                              const float* __restrict__ ur, const float* __restrict__ ui,
                                              float* __restrict__ Tr, float* __restrict__ Ti) {
  __shared__ float lAr[256 * APITCH];
  __shared__ float lAi[256 * APITCH];
  const int t    = threadIdx.x;
  const int lane = t & 31;
  const int wave = t >> 5;
  const int blk  = blockIdx.x;
  const int b    = blk >> 9;
  const int ti   = (blk >> 3) & 63;
  const int tjg  = blk & 7;
  const int tj   = tjg * 8 + wave;
  const int m  = lane & 15, n = lane & 15, hi = lane >> 4;
  const int q    = tjg >> 1;                   // column block (channel), uniform per block
  const int jrow = (tj & 15) * 16 + n;         // u row supplying the conj-transposed column
  const float* sr = Sr + (size_t)b * NIN * NIN;
  const float* si = Si + (size_t)b * NIN * NIN;

  // stage the 16-row S tile (rows ti*16.., cols q*256..q*256+255) in LDS
#pragma unroll
  for (int iter = 0; iter < 16; ++iter) {
    lAr[t * APITCH + iter] = sr[(size_t)(ti * 16 + iter) * NIN + q * 256 + t];
    lAi[t * APITCH + iter] = si[(size_t)(ti * 16 + iter) * NIN + q * 256 + t];
  }
  __syncthreads();

  v8f accr = {}, acci = {};
  for (int k = 0; k < 256; k += 4) {
    const int kk = k + 2 * hi;
    v2f ar, ai, nar;
    ar.x = lAr[kk * APITCH + m];       ar.y = lAr[(kk + 1) * APITCH + m];
    ai.x = lAi[kk * APITCH + m];       ai.y = lAi[(kk + 1) * APITCH + m];
    nar.x = -ar.x;                     nar.y = -ar.y;
    v2f br = *(const v2f*)(ur + jrow * 256 + kk);
    v2f bi = *(const v2f*)(ui + jrow * 256 + kk);
    accr = __builtin_amdgcn_wmma_f32_16x16x4_f32(false, ar,  false, br, (short)0, accr, false, false);
    accr = __builtin_amdgcn_wmma_f32_16x16x4_f32(false, ai,  false, bi, (short)0, accr, false, false);
    acci = __builtin_amdgcn_wmma_f32_16x16x4_f32(false, ai,  false, br, (short)0, acci, false, false);
    acci = __builtin_amdgcn_wmma_f32_16x16x4_f32(false, nar, false, bi, (short)0, acci, false, false);
  }
  float* tr  = Tr + (size_t)b * NIN * NIN;
  float* tii = Ti + (size_t)b * NIN * NIN;
  const int rbase = ti * 16 + 8 * hi;
  const int col   = tj * 16 + n;
#pragma unroll
  for (int v = 0; v < 8; ++v) {
    tr[(size_t)(rbase + v) * NIN + col]  = accr[v];
    tii[(size_t)(rbase + v) * NIN + col] = acci[v];
  }
}

// ---------------- Stage C: out[(a,i),(c,j)] = sum_{p,q} w[a,c,p,q] * T[(p,i),(q,j)] ----------------
// Memory-bound: T stays L2-resident (8MB/batch vs 192MB L2); the 256MB output is a
// write-once stream -> non-temporal stores.
__global__ __launch_bounds__(256) void stageC(const float* __restrict__ Tr, const float* __restrict__ Ti,
                                              const float* __restrict__ Wr, const float* __restrict__ Wi,
                                              float* __restrict__ out) {
  const size_t idx = (size_t)blockIdx.x * 256 + threadIdx.x;   // 8*2048*2048 elements
  const int col = (int)(idx & 2047);
  const int row = (int)((idx >> 11) & 2047);
  const int b   = (int)(idx >> 22);
  const int a = row >> 8, i = row & 255;
  const int c = col >> 8, j = col & 255;
  const float* tr = Tr + (size_t)b * NIN * NIN;
  const float* ti = Ti + (size_t)b * NIN * NIN;
  const float* wr = Wr + ((a * 8 + c) * 16);   // uniform per block (col blocks are 256-aligned)
  const float* wi = Wi + ((a * 8 + c) * 16);
  float accr = 0.f, acci = 0.f;
#pragma unroll
  for (int p = 0; p < 4; ++p)
#pragma unroll
    for (int q = 0; q < 4; ++q) {
      float xr = tr[(size_t)(p * 256 + i) * NIN + q * 256 + j];
      float xi = ti[(size_t)(p * 256 + i) * NIN + q * 256 + j];
      float wre = wr[p * 4 + q], wim = wi[p * 4 + q];
      accr = fmaf(wre, xr, fmaf(-wim, xi, accr));
      acci = fmaf(wre, xi, fmaf(wim, xr, acci));
    }
  __builtin_nontemporal_store(accr, &out[idx * 2]);
  __builtin_nontemporal_store(acci, &out[idx * 2 + 1]);
}

extern "C" void kernel_launch(void* const* d_in, const int* in_sizes, int n_in,
                              void* d_out, int out_size, void* d_ws, size_t ws_size,
                              hipStream_t stream) {
  const float* rho_real = (const float*)d_in[0];
  const float* rho_imag = (const float*)d_in[1];
  const float* phi      = (const float*)d_in[2];
  float* ws = (float*)d_ws;
  // workspace layout (floats)
  float* ur = ws;                 // 65536
  float* ui = ws + 65536;         // 65536
  float* wr = ws + 131072;        // 1024
  float* wi = ws + 132096;        // 1024
  float* Sr = ws + 262144;        // 8*1024*1024 each for Sr,Si,Tr,Ti
  float* Si = Sr + (size_t)BATCH * NIN * NIN;
  float* Tr = Si + (size_t)BATCH * NIN * NIN;
  float* Ti = Tr + (size_t)BATCH * NIN * NIN;
  float* out = (float*)d_out;

  hipLaunchKernelGGL(stage0, dim3(1),      dim3(256), 0, stream, phi, ur, ui, wr, wi);
  hipLaunchKernelGGL(stageA, dim3(4096),   dim3(256), 0, stream, rho_real, rho_imag, ur, ui, Sr, Si);
  hipLaunchKernelGGL(stageB, dim3(4096),   dim3(256), 0, stream, Sr, Si, ur, ui, Tr, Ti);
  hipLaunchKernelGGL(stageC, dim3(131072), dim3(256), 0, stream, Tr, Ti, wr, wi, out);
}